// MPNNLayer_69320772158259
// MI455X (gfx1250) — compile-verified
//
#include <hip/hip_runtime.h>

#define DIM 128

typedef __attribute__((ext_vector_type(2))) float v2f;
typedef __attribute__((ext_vector_type(8))) float v8f;

#if defined(__HIP_PLATFORM_AMD__)
#define fatomicAdd unsafeAtomicAdd
#else
#define fatomicAdd atomicAdd
#endif

// ---------------------------------------------------------------------------
// Kernel 1: compose the two linear layers: WcT[k][n] = (W_gcn @ W_lin)[n][k]
// h = (x W_lin^T) W_gcn^T = x (W_gcn W_lin)^T  -> store transposed for GEMM B.
// ---------------------------------------------------------------------------
__global__ void wct_kernel(const float* __restrict__ W_lin,
                           const float* __restrict__ W_gcn,
                           float* __restrict__ wct) {
    const int n = blockIdx.x;    // 0..127
    const int k = threadIdx.x;   // 0..127
    float s = 0.0f;
    for (int j = 0; j < DIM; ++j)
        s += W_gcn[n * DIM + j] * W_lin[j * DIM + k];
    wct[k * DIM + n] = s;        // WcT is [K][N]
}

// ---------------------------------------------------------------------------
// Kernel 2: h = x @ WcT using V_WMMA_F32_16X16X4_F32.
// 256 threads = 8 waves; each wave computes a 16(M)x128(N) tile of h.
// LDS holds WcT K-pair-packed: ldsp[k/2][n] = {WcT[k][n], WcT[k+1][n]}
// so each WMMA B fragment is ONE aligned ds_load_b64 (no VGPR repacking).
// ---------------------------------------------------------------------------
__global__ void __launch_bounds__(256)
gemm_wmma_f32(const float* __restrict__ x, const float* __restrict__ wct,
              float* __restrict__ h, int N) {
    __shared__ float2 ldsp[(DIM / 2) * DIM];   // 64 KB

    const int tid = threadIdx.x;
    // cooperative fill with K-pair packing; coalesced over n
    for (int i = tid; i < (DIM / 2) * DIM; i += 256) {
        const int kp = i >> 7;          // K pair index 0..63
        const int n  = i & (DIM - 1);   // column 0..127
        float2 p;
        p.x = wct[(2 * kp) * DIM + n];
        p.y = wct[(2 * kp + 1) * DIM + n];
        ldsp[kp * DIM + n] = p;
    }
    __syncthreads();

    const int lane = tid & 31;
    const int wave = tid >> 5;
    const int m    = lane & 15;       // row/col within 16-wide tile
    const int hi   = lane >> 4;       // 0: K offset 0/1, 1: K offset 2/3
    const int row_tile = blockIdx.x * 8 + wave;

    int row = row_tile * 16 + m;
    const int row_clamped = row < N ? row : (N - 1);
    const float2* xr2 = (const float2*)(x + (size_t)row_clamped * DIM);

    v8f acc[8] = {};                  // 8 N-tiles of 16x16 fp32 accumulators

    for (int k = 0; k < DIM; k += 4) {
        // A fragment: lane holds x[row][k + 2*hi], x[row][k + 2*hi + 1]
        const float2 a2 = xr2[(k >> 1) + hi];
        v2f A; A.x = a2.x; A.y = a2.y;
        const int kp = (k >> 1) + hi;   // packed K-pair row in LDS
#pragma unroll
        for (int nt = 0; nt < 8; ++nt) {
            const float2 b2 = ldsp[kp * DIM + nt * 16 + m];
            v2f B; B.x = b2.x; B.y = b2.y;
            acc[nt] = __builtin_amdgcn_wmma_f32_16x16x4_f32(
                false, A, false, B, (short)0, acc[nt], false, false);
        }
    }

    // Store: C/D layout -> VGPR v, lane l: row = v + 8*hi, col = l%16
#pragma unroll
    for (int v = 0; v < 8; ++v) {
        const int row_out = row_tile * 16 + v + hi * 8;
        if (row_out < N) {
            float* hrow = h + (size_t)row_out * DIM;
#pragma unroll
            for (int nt = 0; nt < 8; ++nt) {
                hrow[nt * 16 + m] = acc[nt][v];
            }
        }
    }
}

// ---------------------------------------------------------------------------
// Degree / normalization kernels (deg includes self-loop: +1 per node).
// ---------------------------------------------------------------------------
__global__ void zero_kernel(float* __restrict__ p, int n) {
    const int i = blockIdx.x * blockDim.x + threadIdx.x;
    if (i < n) p[i] = 0.0f;
}

__global__ void deg_kernel(const int* __restrict__ col, float* __restrict__ deg,
                           int E) {
    const int e = blockIdx.x * blockDim.x + threadIdx.x;
    if (e < E) fatomicAdd(&deg[col[e]], 1.0f);
}

__global__ void dinv_kernel(float* __restrict__ deg, int N) {
    const int i = blockIdx.x * blockDim.x + threadIdx.x;
    if (i < N) {
        const float d = deg[i] + 1.0f;   // + self loop; always > 0
        deg[i] = rsqrtf(d);
    }
}

// ---------------------------------------------------------------------------
// out[i][d] = b_gcn[d] + h[i][d] * dinv[i]^2   (bias + self-loop message)
// ---------------------------------------------------------------------------
__global__ void out_init_kernel(const float* __restrict__ h,
                                const float* __restrict__ dinv,
                                const float* __restrict__ b,
                                float* __restrict__ out, int N) {
    const int idx = blockIdx.x * blockDim.x + threadIdx.x;
    if (idx < N * DIM) {
        const int i = idx >> 7;
        const int d = idx & (DIM - 1);
        const float di = dinv[i];
        out[idx] = b[d] + h[idx] * di * di;
    }
}

// ---------------------------------------------------------------------------
// Edge scatter: one wave32 per edge. Lane loads 16B of h[row], emits 4
// native fp32 atomic adds into out[col]. h (51.2 MB) is L2-resident.
// ---------------------------------------------------------------------------
__global__ void __launch_bounds__(256)
edge_scatter(const int* __restrict__ edges, const float* __restrict__ h,
             const float* __restrict__ dinv, float* __restrict__ out, int E) {
    const int gid  = blockIdx.x * blockDim.x + threadIdx.x;
    const int e    = gid >> 5;
    const int lane = gid & 31;
    if (e >= E) return;

    const int r = edges[e];         // source
    const int c = edges[E + e];     // target
    const float w = dinv[r] * dinv[c];

    const float4 v = ((const float4*)(h + (size_t)r * DIM))[lane];
    float* orow = out + (size_t)c * DIM + lane * 4;
    fatomicAdd(orow + 0, v.x * w);
    fatomicAdd(orow + 1, v.y * w);
    fatomicAdd(orow + 2, v.z * w);
    fatomicAdd(orow + 3, v.w * w);
}

// ---------------------------------------------------------------------------
extern "C" void kernel_launch(void* const* d_in, const int* in_sizes, int n_in,
                              void* d_out, int out_size, void* d_ws, size_t ws_size,
                              hipStream_t stream) {
    const float* x     = (const float*)d_in[0];
    const int*   edges = (const int*)d_in[1];     // [2, E] flat, int32 (JAX x64 off)
    const float* W_lin = (const float*)d_in[2];
    const float* W_gcn = (const float*)d_in[3];
    const float* b_gcn = (const float*)d_in[4];

    const int N = in_sizes[0] / DIM;
    const int E = in_sizes[1] / 2;
    float* out = (float*)d_out;

    // workspace: h [N*DIM] | dinv [N] | WcT [DIM*DIM]  (~51.7 MB total)
    float* h    = (float*)d_ws;
    float* dinv = h + (size_t)N * DIM;
    float* wct  = dinv + N;

    // weight composition (tiny)
    wct_kernel<<<DIM, DIM, 0, stream>>>(W_lin, W_gcn, wct);

    // degree -> dinv
    zero_kernel<<<(N + 255) / 256, 256, 0, stream>>>(dinv, N);
    deg_kernel<<<(E + 255) / 256, 256, 0, stream>>>(edges + E, dinv, E);
    dinv_kernel<<<(N + 255) / 256, 256, 0, stream>>>(dinv, N);

    // h = x @ WcT via WMMA fp32 (8 waves/block, 128 rows/block)
    gemm_wmma_f32<<<(N + 127) / 128, 256, 0, stream>>>(x, wct, h, N);

    // out = bias + self-loop term
    out_init_kernel<<<(N * DIM + 255) / 256, 256, 0, stream>>>(h, dinv, b_gcn, out, N);

    // edge messages: wave-per-edge scatter-add
    const long long scatter_threads = (long long)E * 32;
    edge_scatter<<<(unsigned)((scatter_threads + 255) / 256), 256, 0, stream>>>(
        edges, h, dinv, out, E);
}